// DSnetwork_59004260712466
// MI455X (gfx1250) — compile-verified
//
#include <hip/hip_runtime.h>

#define S_SEG 131072
#define G_SEG 4096
#define D 128

typedef __attribute__((ext_vector_type(2))) float v2f;
typedef __attribute__((ext_vector_type(8))) float v8f;

enum { ACT_NONE = 0, ACT_ELU = 1, ACT_RELU = 2 };

// ---------------------------------------------------------------------------
// Segment bounds via binary search on a sorted index array.
// bounds[s] = lower_bound(idx, s); launched with nseg+1 threads so
// bounds[nseg] == n.
// ---------------------------------------------------------------------------
__global__ void bounds_kernel(const int* __restrict__ idx, int n, int nseg,
                              int* __restrict__ bounds) {
  int s = blockIdx.x * blockDim.x + threadIdx.x;
  if (s > nseg) return;
  int lo = 0, hi = n;
  while (lo < hi) {
    int mid = (lo + hi) >> 1;
    if (idx[mid] < s) lo = mid + 1; else hi = mid;
  }
  bounds[s] = lo;
}

// ---------------------------------------------------------------------------
// Sorted segment mean: one 128-thread block per segment, thread = channel.
// Rows of a segment are contiguous -> fully coalesced 512B row reads,
// zero atomics. Empty segments write 0 (matches s / max(c,1)).
// ---------------------------------------------------------------------------
__global__ __launch_bounds__(128) void seg_mean_kernel(
    const float* __restrict__ x, const int* __restrict__ bounds,
    float* __restrict__ out) {
  int s = blockIdx.x;
  int c = threadIdx.x;
  int beg = bounds[s], end = bounds[s + 1];
  float s0 = 0.f, s1 = 0.f, s2 = 0.f, s3 = 0.f;
  int r = beg;
  for (; r + 4 <= end; r += 4) {
    s0 += x[(size_t)(r + 0) * D + c];
    s1 += x[(size_t)(r + 1) * D + c];
    s2 += x[(size_t)(r + 2) * D + c];
    s3 += x[(size_t)(r + 3) * D + c];
  }
  for (; r < end; ++r) s0 += x[(size_t)r * D + c];
  float sum = (s0 + s1) + (s2 + s3);
  float cnt = (float)(end - beg);
  out[(size_t)s * D + c] = sum / fmaxf(cnt, 1.0f);
}

// ---------------------------------------------------------------------------
// WMMA fp32 GEMM: OUT[M,N] = act(A[M,128] @ W[128,N] + bias
//                                + (GATHER ? addv[gidx[m],N] : 0))
// Block = 256 threads (8 wave32), owns a 16-row M tile; wave owns N/128
// 16-col N-tiles. A tile staged in LDS (stride 132 -> conflict-free column
// reads). Safe to run with OUT == A (all A reads precede the barrier).
// Uses V_WMMA_F32_16X16X4_F32, 32 issues per K=128.
// ---------------------------------------------------------------------------
template <int N, int ACT, bool GATHER>
__global__ __launch_bounds__(256) void gemm_wmma_kernel(
    const float* A, const float* __restrict__ W, const float* __restrict__ bias,
    const float* __restrict__ addv, const int* __restrict__ gidx, float* OUT) {
  __shared__ float lds_a[16 * 132];
  const int tid = threadIdx.x;
  const int tile0 = blockIdx.x * 16;

  // cooperative, coalesced load of the contiguous 16x128 A tile
  const float* Ablk = A + (size_t)tile0 * D;
#pragma unroll
  for (int i = 0; i < (16 * D) / 256; ++i) {
    int e = i * 256 + tid;
    lds_a[(e >> 7) * 132 + (e & 127)] = Ablk[e];
  }
  __syncthreads();

  const int wave = tid >> 5;
  const int lane = tid & 31;
  const int half = lane >> 4;   // lanes 16-31 hold K+2 / M+8 halves
  const int l16  = lane & 15;

#pragma unroll
  for (int t = 0; t < N / 128; ++t) {
    const int nt   = wave + t * 8;      // 16-wide N tile index
    const int ncol = nt * 16 + l16;
    v8f acc = {};
#pragma unroll
    for (int k = 0; k < 128; k += 4) {
      const int ka = k + half * 2;
      v2f a, b;
      a.x = lds_a[l16 * 132 + ka];      // A frag: M=l16, K=ka..ka+1
      a.y = lds_a[l16 * 132 + ka + 1];
      b.x = W[(size_t)ka * N + ncol];   // B frag: K=ka..ka+1, N=ncol
      b.y = W[(size_t)(ka + 1) * N + ncol];
      acc = __builtin_amdgcn_wmma_f32_16x16x4_f32(
          false, a, false, b, (short)0, acc, false, false);
    }
    const float bn = bias[ncol];
#pragma unroll
    for (int v = 0; v < 8; ++v) {
      const int m = tile0 + half * 8 + v;   // C/D layout: VGPR v -> M=v(+8)
      float x = acc[v] + bn;
      if (GATHER) x += addv[(size_t)gidx[m] * N + ncol];
      if (ACT == ACT_ELU)       x = x > 0.f ? x : expm1f(x);
      else if (ACT == ACT_RELU) x = fmaxf(x, 0.f);
      OUT[(size_t)m * N + ncol] = x;
    }
  }
}

// ---------------------------------------------------------------------------
// Tiny tail: out[g,j] = t[g,:256] . Wf2[:,j] + bf2[j]   (4096 x 10)
// ---------------------------------------------------------------------------
__global__ void final_kernel(const float* __restrict__ t,
                             const float* __restrict__ Wf2,
                             const float* __restrict__ bf2,
                             float* __restrict__ out) {
  int i = blockIdx.x * blockDim.x + threadIdx.x;
  if (i >= G_SEG * 10) return;
  int g = i / 10, j = i % 10;
  float s = bf2[j];
#pragma unroll 8
  for (int k = 0; k < 256; ++k) s += t[(size_t)g * 256 + k] * Wf2[k * 10 + j];
  out[i] = s;
}

extern "C" void kernel_launch(void* const* d_in, const int* in_sizes, int n_in,
                              void* d_out, int out_size, void* d_ws, size_t ws_size,
                              hipStream_t stream) {
  (void)n_in; (void)out_size; (void)ws_size;
  const float* h_node  = (const float*)d_in[0];
  const float* W_fc0   = (const float*)d_in[1];
  const float* b_fc0   = (const float*)d_in[2];
  const float* W_sum0  = (const float*)d_in[3];
  const float* b_sum0  = (const float*)d_in[4];
  const float* W_fc1   = (const float*)d_in[5];
  const float* b_fc1   = (const float*)d_in[6];
  const float* W_sum1  = (const float*)d_in[7];
  const float* b_sum1  = (const float*)d_in[8];
  const float* Wf1     = (const float*)d_in[9];
  const float* bf1     = (const float*)d_in[10];
  const float* Wf2     = (const float*)d_in[11];
  const float* bf2     = (const float*)d_in[12];
  const int*   sub_batch = (const int*)d_in[13];   // [N], sorted
  const int*   sub_idx   = (const int*)d_in[14];   // [S], sorted
  const int N = in_sizes[0] / D;                   // 2,000,000

  // workspace layout (all offsets 256B-aligned by construction)
  char* ws = (char*)d_ws;
  size_t o = 0;
  float* h_sub  = (float*)(ws + o); o += (size_t)S_SEG * D * 4;   // 64 MB
  float* g_mean = (float*)(ws + o); o += (size_t)G_SEG * D * 4;   //  2 MB
  float* x2     = (float*)(ws + o); o += (size_t)G_SEG * D * 4;   //  2 MB
  float* tbuf   = (float*)(ws + o); o += (size_t)G_SEG * 256 * 4; //  4 MB
  int* bounds_sub = (int*)(ws + o); o += ((size_t)(S_SEG + 1) * 4 + 255) & ~(size_t)255;
  int* bounds_grp = (int*)(ws + o); o += ((size_t)(G_SEG + 1) * 4 + 255) & ~(size_t)255;

  // segment bounds (sorted indices -> binary search, no atomics anywhere)
  bounds_kernel<<<(S_SEG + 256) / 256, 256, 0, stream>>>(sub_batch, N, S_SEG, bounds_sub);
  bounds_kernel<<<(G_SEG + 256) / 256, 256, 0, stream>>>(sub_idx, S_SEG, G_SEG, bounds_grp);

  // subgraph pooling: [N,128] -> [S,128]   (the 1 GB memory-bound pass)
  seg_mean_kernel<<<S_SEG, 128, 0, stream>>>(h_node, bounds_sub, h_sub);

  // DS layer 0
  seg_mean_kernel<<<G_SEG, 128, 0, stream>>>(h_sub, bounds_grp, g_mean);
  gemm_wmma_kernel<128, ACT_NONE, false><<<G_SEG / 16, 256, 0, stream>>>(
      g_mean, W_sum0, b_sum0, nullptr, nullptr, x2);
  gemm_wmma_kernel<128, ACT_ELU, true><<<S_SEG / 16, 256, 0, stream>>>(
      h_sub, W_fc0, b_fc0, x2, sub_idx, h_sub);          // in place

  // DS layer 1
  seg_mean_kernel<<<G_SEG, 128, 0, stream>>>(h_sub, bounds_grp, g_mean);
  gemm_wmma_kernel<128, ACT_NONE, false><<<G_SEG / 16, 256, 0, stream>>>(
      g_mean, W_sum1, b_sum1, nullptr, nullptr, x2);
  gemm_wmma_kernel<128, ACT_ELU, true><<<S_SEG / 16, 256, 0, stream>>>(
      h_sub, W_fc1, b_fc1, x2, sub_idx, h_sub);          // in place

  // graph pooling + final MLP
  seg_mean_kernel<<<G_SEG, 128, 0, stream>>>(h_sub, bounds_grp, g_mean);
  gemm_wmma_kernel<256, ACT_RELU, false><<<G_SEG / 16, 256, 0, stream>>>(
      g_mean, Wf1, bf1, nullptr, nullptr, tbuf);
  final_kernel<<<(G_SEG * 10 + 255) / 256, 256, 0, stream>>>(tbuf, Wf2, bf2, (float*)d_out);
}